// DRMMLogCountHistogram_24816321036869
// MI455X (gfx1250) — compile-verified
//
#include <hip/hip_runtime.h>
#include <hip/hip_bf16.h>
#include <cstdint>
#include <cstddef>

// DRMM log-count histogram for MI455X (gfx1250).
// B=64, C=4, Q=32, D=4096, NBINS=30.
// Memory-bound: 128 MiB simmat stream -> ~5.6us at 23.3 TB/s.
// Strategy: async global->LDS streaming (ASYNCcnt), lane-privatized
// conflict-free LDS histograms, block-shared dtoks validity bitmask.

#define NBINS   30
#define B_      64
#define C_      4
#define Q_      32
#define D_      4096
#define WPB     8              // waves per block (256 threads)
#define CHUNK   128            // floats per chunk per wave (32 lanes x 16B)
#define NCHUNK  (D_ / CHUNK)   // 32 chunks per row
#define DEPTH   4              // async ring depth

// ---- async global -> LDS copy, 16B per lane (CDNA5 async-tensor path) ----
__device__ __forceinline__ void async_copy_b128(uint32_t lds_byte_addr,
                                                const void* global_addr) {
  asm volatile("global_load_async_to_lds_b128 %0, %1, off"
               :: "v"(lds_byte_addr), "v"(global_addr)
               : "memory");
}

__device__ __forceinline__ void compiler_mem_fence() {
  asm volatile("" ::: "memory");
}

__global__ __launch_bounds__(256)
void drmm_hist_kernel(const float* __restrict__ simmat,
                      const int*  __restrict__ dtoks,
                      const int*  __restrict__ qtoks,
                      float* __restrict__ out)
{
  __shared__ float    buf[WPB][DEPTH][CHUNK];     // 16 KiB async staging
  __shared__ uint32_t hist[WPB][NBINS][32];       // 30 KiB lane-private hists
  __shared__ uint32_t dmask[D_ / 32];             // 512 B validity bits

  const int tid  = (int)threadIdx.x;
  const int w    = tid >> 5;
  const int lane = tid & 31;
  const int row0 = (int)blockIdx.x * WPB;
  const int b    = row0 >> 7;                      // C_*Q_ == 128; shared by block
  const int row  = row0 + w;                       // flat (b,c,q) row index
  const int q    = row & (Q_ - 1);

  // ---- build dtoks!=-1 bitmask for this block's batch b (once) ----
  if (tid < D_ / 32) {
    const int dbase = tid * 32;
    const int4* dp = (const int4*)(dtoks + (size_t)b * D_ + dbase);
    uint32_t m = 0u;
#pragma unroll
    for (int g = 0; g < 8; ++g) {
      int4 v = dp[g];
      m |= (uint32_t)(v.x != -1) << (g * 4 + 0);
      m |= (uint32_t)(v.y != -1) << (g * 4 + 1);
      m |= (uint32_t)(v.z != -1) << (g * 4 + 2);
      m |= (uint32_t)(v.w != -1) << (g * 4 + 3);
    }
    dmask[tid] = m;
  }
  __syncthreads();

  float* orow = out + (size_t)row * NBINS;
  const int qt = qtoks[(size_t)b * Q_ + q];
  if (qt == -1) {
    // all weights zero -> hist == 0 -> log(1e-5)
    if (lane < NBINS) orow[lane] = -11.512925465f;
    return;
  }

  // ---- zero this wave's private histogram (bank-conflict-free layout) ----
  uint32_t* hw = &hist[w][0][0];
#pragma unroll
  for (int j = 0; j < NBINS; ++j) hw[j * 32 + lane] = 0u;

  const float*  srow    = simmat + (size_t)row * D_;
  const uint32_t ldsbase = (uint32_t)(uintptr_t)&buf[w][0][0];

  auto issue = [&](int t) {
    uint32_t dst = ldsbase + (uint32_t)(((t & (DEPTH - 1)) * CHUNK + lane * 4) * 4);
    async_copy_b128(dst, (const void*)(srow + (size_t)t * CHUNK + lane * 4));
  };

  auto process = [&](int t) {
    const float4 v = *(const float4*)&buf[w][t & (DEPTH - 1)][lane * 4];
    // 4 consecutive d-values per lane: d = t*128 + lane*4 .. +3
    const uint32_t md   = dmask[t * 4 + (lane >> 3)];
    const uint32_t bits = (md >> ((lane & 7) * 4)) & 0xFu;
    const float s[4] = {v.x, v.y, v.z, v.w};
#pragma unroll
    for (int e = 0; e < 4; ++e) {
      if (bits & (1u << e)) {
        int bin = (int)(((s[e] + 1.000001f) * 0.5f) * 29.0f);
        bin = bin < 0 ? 0 : (bin > NBINS - 1 ? NBINS - 1 : bin);
        hw[bin * 32 + lane] += 1u;   // conflict-free: bank = lane + 32*(bin&1)
      }
    }
  };

  // ---- prologue: fill DEPTH-1 slots of the ring ----
#pragma unroll
  for (int p = 0; p < DEPTH - 1; ++p) issue(p);

  // ---- steady state: issue t+3, wait oldest, consume t ----
  for (int t = 0; t <= NCHUNK - DEPTH; ++t) {
    issue(t + DEPTH - 1);
    __builtin_amdgcn_s_wait_asynccnt(DEPTH - 1);
    compiler_mem_fence();
    process(t);
  }
  // ---- tail: drain with tightening constant-immediate waits ----
  __builtin_amdgcn_s_wait_asynccnt(2); compiler_mem_fence(); process(NCHUNK - 3);
  __builtin_amdgcn_s_wait_asynccnt(1); compiler_mem_fence(); process(NCHUNK - 2);
  __builtin_amdgcn_s_wait_asynccnt(0); compiler_mem_fence(); process(NCHUNK - 1);

  // ---- reduce 32 lane columns per bin, write log(hist + 1e-5) ----
  if (lane < NBINS) {
    uint32_t s = 0u;
#pragma unroll
    for (int k = 0; k < 32; ++k) s += hw[lane * 32 + k];
    orow[lane] = logf((float)s + 1e-5f);
  }
}

extern "C" void kernel_launch(void* const* d_in, const int* in_sizes, int n_in,
                              void* d_out, int out_size, void* d_ws, size_t ws_size,
                              hipStream_t stream) {
  (void)in_sizes; (void)n_in; (void)d_ws; (void)ws_size; (void)out_size;
  const float* simmat = (const float*)d_in[0];
  const int*   dtoks  = (const int*)d_in[1];
  const int*   qtoks  = (const int*)d_in[2];
  float*       out    = (float*)d_out;

  const int rows   = B_ * C_ * Q_;          // 8192
  const int blocks = rows / WPB;            // 1024
  drmm_hist_kernel<<<dim3(blocks), dim3(256), 0, stream>>>(simmat, dtoks, qtoks, out);
}